// Plane_Height_loss_45062796869891
// MI455X (gfx1250) — compile-verified
//
#include <hip/hip_runtime.h>
#include <math.h>

typedef float v2f __attribute__((ext_vector_type(2)));
typedef float v8f __attribute__((ext_vector_type(8)));

#define B_   32
#define H_   384
#define W_   1280
#define OH   96
#define OW   320
#define NSAMP (OH*OW)        // 30720 samples per batch
#define NBLK  15             // blocks per batch
#define SPB   (NSAMP/NBLK)   // 2048 samples per block
#define TPB   256
#define SPT   (SPB/TPB)      // 8 samples per thread

// d_ws float layout
#define WS_KINV 0                  // 32 * 12 floats
#define WS_PART 512                // 32 * 15 * 16 floats
#define WS_LOSS (512 + 32*15*16)   // 32 floats

// ---------------- kernel 0: per-batch inverse of scaled intrinsics ----------------
__global__ void k0_kinv(const float* __restrict__ intr, float* ws) {
    int b = threadIdx.x;
    if (b >= B_) return;
    const float* K = intr + b * 9;
    // rows 0 and 1 scaled by 1/4 (downsample factor)
    float a00 = K[0]*0.25f, a01 = K[1]*0.25f, a02 = K[2]*0.25f;
    float a10 = K[3]*0.25f, a11 = K[4]*0.25f, a12 = K[5]*0.25f;
    float a20 = K[6],       a21 = K[7],       a22 = K[8];
    float det = a00*(a11*a22 - a12*a21) - a01*(a10*a22 - a12*a20) + a02*(a10*a21 - a11*a20);
    float id  = 1.0f / det;
    float* o = ws + WS_KINV + b * 12;
    o[0] = (a11*a22 - a12*a21)*id; o[1] = (a02*a21 - a01*a22)*id; o[2] = (a01*a12 - a02*a11)*id;
    o[3] = (a12*a20 - a10*a22)*id; o[4] = (a00*a22 - a02*a20)*id; o[5] = (a02*a10 - a00*a12)*id;
    o[6] = (a10*a21 - a11*a20)*id; o[7] = (a01*a20 - a00*a21)*id; o[8] = (a00*a11 - a01*a10)*id;
}

// ---------------- kernel 1: weighted moment accumulation + WMMA reduction ----------------
__global__ void __launch_bounds__(TPB)
k1_moments(const float* __restrict__ plane, const float* __restrict__ depth,
           float* ws) {
    const int t     = threadIdx.x;
    const int batch = blockIdx.y;

    const float* Ki = ws + WS_KINV + batch * 12;
    const float k00 = Ki[0], k01 = Ki[1], k02 = Ki[2];
    const float k10 = Ki[3], k11 = Ki[4], k12 = Ki[5];
    const float k20 = Ki[6], k21 = Ki[7], k22 = Ki[8];

    const size_t boff = (size_t)batch * ((size_t)H_ * W_);
    const float* P = plane + boff;
    const float* D = depth + boff;

    float s0=0.f,s1=0.f,s2=0.f,s3=0.f,s4=0.f,s5=0.f,s6=0.f,s7=0.f,s8=0.f,s9=0.f;
    const int fbase = blockIdx.x * SPB;

#pragma unroll
    for (int k = 0; k < SPT; ++k) {
        int f  = fbase + k * TPB + t;          // always < 30720 (15*2048 exact)
        int i  = f / OW;
        int j  = f - i * OW;
        int r0 = 4*i + 1;
        int cb = 4*j;                           // 16B-aligned float4 covering cols 4j..4j+3
        const float4 p0 = *(const float4*)(P + (size_t)r0    * W_ + cb);
        const float4 p1 = *(const float4*)(P + (size_t)(r0+1)* W_ + cb);
        const float4 d0 = *(const float4*)(D + (size_t)r0    * W_ + cb);
        const float4 d1 = *(const float4*)(D + (size_t)(r0+1)* W_ + cb);
        // bilinear /4: mean of 2x2 block at cols 4j+1,4j+2 (elements .y,.z); plane is cubed first
        float w = 0.25f * (p0.y*p0.y*p0.y + p0.z*p0.z*p0.z + p1.y*p1.y*p1.y + p1.z*p1.z*p1.z);
        float d = 0.25f * (d0.y + d0.z + d1.y + d1.z);
        float fj = (float)j, fi = (float)i;
        float x = (k00*fj + k01*fi + k02) * d;  // cam = Kinv * (j,i,1) * depth
        float y = (k10*fj + k11*fi + k12) * d;
        float z = (k20*fj + k21*fi + k22) * d;
        float wx = w*x, wy = w*y, wz = w*z;
        s0 += w;    s1 += wx;   s2 += wy;   s3 += wz;
        s4 += wx*x; s5 += wx*y; s6 += wx*z;
        s7 += wy*y; s8 += wy*z; s9 += wz*z;
    }

    // ---- block reduction of 10 partials via V_WMMA_F32_16X16X4_F32, A = ones ----
    // With A[m][k] == 1 for all slots (layout-independent), D[m][n] = sum_k B[k][n]:
    // one WMMA sums the 4 B-register values of lane pair (n, n+16) and broadcasts
    // the column sum to every lane's d[0].
    __shared__ float Wq[10 * 257];   // quantity-major, stride 257 => conflict-free reads
    __shared__ float Pls[8 * 16];    // per-wave partials

    Wq[0*257+t]=s0; Wq[1*257+t]=s1; Wq[2*257+t]=s2; Wq[3*257+t]=s3; Wq[4*257+t]=s4;
    Wq[5*257+t]=s5; Wq[6*257+t]=s6; Wq[7*257+t]=s7; Wq[8*257+t]=s8; Wq[9*257+t]=s9;
    __syncthreads();

    const int lane = t & 31, wv = t >> 5;
    const int c = lane & 15, half = lane >> 4;
    const int qq   = (c < 10) ? c : (c - 10);   // dup real data into cols 10-15 (ignored)
    const int base = wv * 32 + half * 16;
    float b0 = 0.f, b1 = 0.f;
#pragma unroll
    for (int jj = 0; jj < 8; ++jj) {
        b0 += Wq[qq*257 + base + jj];
        b1 += Wq[qq*257 + base + 8 + jj];
    }
    v2f ones; ones[0] = 1.0f; ones[1] = 1.0f;
    v2f bv;   bv[0] = b0;     bv[1] = b1;
    v8f acc = {0.f,0.f,0.f,0.f,0.f,0.f,0.f,0.f};
    acc = __builtin_amdgcn_wmma_f32_16x16x4_f32(false, ones, false, bv,
                                                (short)0, acc, false, false);
    // acc[0] on every lane = this wave's 32-thread sum of quantity (lane%16)
    if (half == 0) Pls[wv*16 + c] = (c < 10) ? acc[0] : 0.0f;
    __syncthreads();

    if (wv == 0) {  // wave-uniform branch: EXEC all ones inside
        v2f bA; bA[0] = Pls[(2*half + 0)*16 + c]; bA[1] = Pls[(2*half + 1)*16 + c];
        v2f bB; bB[0] = Pls[(2*half + 4)*16 + c]; bB[1] = Pls[(2*half + 5)*16 + c];
        v8f d2 = {0.f,0.f,0.f,0.f,0.f,0.f,0.f,0.f};
        d2 = __builtin_amdgcn_wmma_f32_16x16x4_f32(false, ones, false, bA,
                                                   (short)0, d2, false, false);
        d2 = __builtin_amdgcn_wmma_f32_16x16x4_f32(false, ones, false, bB,
                                                   (short)0, d2, false, false);
        if (half == 0 && c < 10)
            ws[WS_PART + ((size_t)batch * NBLK + blockIdx.x) * 16 + c] = d2[0];
    }
}

// ---------------- kernel 2: per-batch solve ----------------
__global__ void k2_solve(float* ws, float* __restrict__ out) {
    const int batch = blockIdx.x;
    const int t = threadIdx.x;
    __shared__ float sP[160];
    __shared__ float SS[10];
    float v = 0.f;
    if (t < NBLK * 10) {
        int blk = t / 10, q = t - blk * 10;
        v = ws[WS_PART + ((size_t)batch * NBLK + blk) * 16 + q];
    }
    sP[t] = v;
    __syncthreads();
    if (t < 10) {
        float a = 0.f;
        for (int blk = 0; blk < NBLK; ++blk) a += sP[blk*10 + t];
        SS[t] = a;
    }
    __syncthreads();
    if (t == 0) {
        float Sw = SS[0], Sx = SS[1], Sy = SS[2], Sz = SS[3];
        float Axx=SS[4], Axy=SS[5], Axz=SS[6], Ayy=SS[7], Ayz=SS[8], Azz=SS[9];
        // symmetric 3x3 inverse via adjugate (pinv == inv, full rank)
        float c00 = Ayy*Azz - Ayz*Ayz;
        float c01 = Axz*Ayz - Axy*Azz;
        float c02 = Axy*Ayz - Axz*Ayy;
        float c11 = Axx*Azz - Axz*Axz;
        float c12 = Axy*Axz - Axx*Ayz;
        float c22 = Axx*Ayy - Axy*Axy;
        float det = Axx*c00 + Axy*c01 + Axz*c02;
        float id  = 1.0f / det;
        float nx = (Sx*c00 + Sy*c01 + Sz*c02) * id;   // normal = left * inv(A)
        float ny = (Sx*c01 + Sy*c11 + Sz*c12) * id;
        float nz = (Sx*c02 + Sy*c12 + Sz*c22) * id;
        float nn = sqrtf(nx*nx + ny*ny + nz*nz);
        // height = (left . n_hat) / Sw   (since sum w*(cam.n) = left.n)
        float height = (Sx*nx + Sy*ny + Sz*nz) / (nn * Sw);
        float scale  = (1.7f / 30.0f) / height;
        out[batch] = scale;
        // |d*s - d| / (d*s) == |s-1|/s elementwise (d > 0): exact per-batch loss
        ws[WS_LOSS + batch] = fabsf(scale - 1.0f) / scale;
    }
}

// ---------------- kernel 3: mean loss ----------------
__global__ void k3_loss(const float* __restrict__ ws, float* __restrict__ out) {
    if (threadIdx.x == 0) {
        float a = 0.f;
        for (int b = 0; b < B_; ++b) a += ws[WS_LOSS + b];
        out[B_] = a * (1.0f / B_);
    }
}

extern "C" void kernel_launch(void* const* d_in, const int* in_sizes, int n_in,
                              void* d_out, int out_size, void* d_ws, size_t ws_size,
                              hipStream_t stream) {
    (void)in_sizes; (void)n_in; (void)out_size; (void)ws_size;
    const float* plane = (const float*)d_in[0];
    const float* depth = (const float*)d_in[1];
    const float* intr  = (const float*)d_in[2];
    float* out = (float*)d_out;
    float* ws  = (float*)d_ws;

    k0_kinv   <<<1, 32, 0, stream>>>(intr, ws);
    k1_moments<<<dim3(NBLK, B_), TPB, 0, stream>>>(plane, depth, ws);
    k2_solve  <<<B_, 160, 0, stream>>>(ws, out);
    k3_loss   <<<1, 32, 0, stream>>>(ws, out);
}